// HGCNADP_84980222918805
// MI455X (gfx1250) — compile-verified
//
#include <hip/hip_runtime.h>
#include <math.h>

typedef __attribute__((ext_vector_type(2))) float v2f;
typedef __attribute__((ext_vector_type(8))) float v8f;

#define Bsz 8
#define Nn  10000
#define Mm  4000
#define Dd  40
#define Kk  20
#define BNK (Bsz * Nn * Kk)
#define NW  4                 // waves per block
#define NTI (Mm / 16)         // 250 column tiles
#define NIT ((NTI + NW - 1) / NW)  // 63 uniform iterations

// Branch-free relu(tanh(2d)) = max(0, (1 - e^-4d) / (1 + e^-4d)).
// d <= 0 -> result 0 (e=inf gives NaN, v_max_num drops NaN for the 0 operand).
__device__ __forceinline__ float relu_tanh2(float d) {
    float e = __expf(-4.0f * d);                       // v_exp_f32 path
    float r = (1.0f - e) * __builtin_amdgcn_rcpf(1.0f + e);
    return fmaxf(r, 0.0f);
}

// Pass 1: DE = tanh(2*nodevec), EE = tanh(2*edgevec) into workspace.
__global__ void hgcn_prep(const float* __restrict__ nodevec,
                          const float* __restrict__ edgevec,
                          float* __restrict__ DE,
                          float* __restrict__ EE) {
    int i = blockIdx.x * blockDim.x + threadIdx.x;
    if (i < Nn * Dd) DE[i] = tanhf(2.0f * nodevec[i]);
    if (i < Mm * Dd) EE[i] = tanhf(2.0f * edgevec[i]);
}

// Pass 2: fused f32-WMMA GEMM + streaming per-row top-20.
// 4 waves per 16-row tile split the 250 column tiles; partial top-20 lists
// merged through LDS at the end. adj is never materialized in HBM.
__launch_bounds__(32 * NW)
__global__ void hgcn_topk(const float* __restrict__ DE,
                          const float* __restrict__ EE,
                          const float* __restrict__ noise,
                          float* __restrict__ out) {
    __shared__ float adjT[NW][16][16];
    __shared__ float nT[NW][16][16];
    __shared__ float mVal[16][NW * Kk];
    __shared__ float mAdj[16][NW * Kk];
    __shared__ int   mCol[16][NW * Kk];

    const int tid     = threadIdx.x;
    const int wv      = tid >> 5;       // wave id within block (uniform per wave)
    const int lane    = tid & 31;
    const int r16     = lane & 15;
    const int rowbase = blockIdx.x * 16;
    const int row     = rowbase + r16;
    const int koff    = (lane >> 4) * 2;  // f32 A/B layout: lanes 0-15 K={0,1}, 16-31 K={2,3}
    const int mofs    = (lane >> 4) * 8;  // f32 C/D layout: lanes 16-31 hold M=g+8

    // Preload A-tile registers for all 10 K-steps (K = 40 = 10 * 4).
    float a0[10], a1[10];
    #pragma unroll
    for (int kk = 0; kk < 10; ++kk) {
        a0[kk] = DE[row * Dd + kk * 4 + koff + 0];
        a1[kk] = DE[row * Dd + kk * 4 + koff + 1];
    }

    // Register-resident partial top-20 (compile-time indexed only).
    float vals[Kk], adjs[Kk];
    int   cols[Kk];
    #pragma unroll
    for (int i = 0; i < Kk; ++i) { vals[i] = -1e30f; adjs[i] = 0.0f; cols[i] = 0; }
    float curmin = -1e30f;
    int   minidx = 0;

    for (int tt = 0; tt < NIT; ++tt) {
        const int  t      = tt * NW + wv;     // wave-uniform
        const bool active = (t < NTI);        // scalar branch: EXEC stays all-ones

        if (active) {
            const int colbase = t * 16;
            const int col     = colbase + r16;

            v8f acc = {};
            #pragma unroll
            for (int kk = 0; kk < 10; ++kk) {
                v2f av; av.x = a0[kk]; av.y = a1[kk];
                v2f bv; bv.x = EE[col * Dd + kk * 4 + koff + 0];
                        bv.y = EE[col * Dd + kk * 4 + koff + 1];
                acc = __builtin_amdgcn_wmma_f32_16x16x4_f32(
                    false, av, false, bv, (short)0, acc, false, false);
            }

            // Stream the matching 16x16 noise tile (read-once -> nontemporal).
            #pragma unroll
            for (int j = 0; j < 8; ++j) {
                int idx = j * 32 + lane;
                int mm = idx >> 4, nn = idx & 15;
                nT[wv][mm][nn] = __builtin_nontemporal_load(
                    &noise[(size_t)(rowbase + mm) * Mm + colbase + nn]);
            }
            // Stage adj = relu(tanh(2*dot)) from the WMMA accumulator layout.
            #pragma unroll
            for (int g = 0; g < 8; ++g)
                adjT[wv][g + mofs][r16] = relu_tanh2(acc[g]);
        }
        __syncthreads();

        if (active && lane < 16) {
            const int colbase = t * 16;
            #pragma unroll
            for (int n = 0; n < 16; ++n) {
                float aj = adjT[wv][lane][n];
                float sc = fmaf(0.01f, nT[wv][lane][n], aj);
                if (sc > curmin) {
                    int c = colbase + n;
                    #pragma unroll
                    for (int i = 0; i < Kk; ++i)
                        if (i == minidx) { vals[i] = sc; adjs[i] = aj; cols[i] = c; }
                    curmin = vals[0]; minidx = 0;
                    #pragma unroll
                    for (int i = 1; i < Kk; ++i)
                        if (vals[i] < curmin) { curmin = vals[i]; minidx = i; }
                }
            }
        }
        __syncthreads();
    }

    // Publish each wave's partial list.
    if (lane < 16) {
        #pragma unroll
        for (int i = 0; i < Kk; ++i) {
            mVal[r16][wv * Kk + i] = vals[i];
            mAdj[r16][wv * Kk + i] = adjs[i];
            mCol[r16][wv * Kk + i] = cols[i];
        }
    }
    __syncthreads();

    // Wave 0 merges the 4 partial lists (80 candidates/row -> top 20).
    if (wv == 0 && lane < 16) {
        #pragma unroll
        for (int i = 0; i < Kk; ++i) { vals[i] = -1e30f; adjs[i] = 0.0f; cols[i] = 0; }
        curmin = -1e30f; minidx = 0;

        for (int i = 0; i < NW * Kk; ++i) {
            float sc = mVal[lane][i];
            if (sc > curmin) {
                float aj = mAdj[lane][i];
                int   c  = mCol[lane][i];
                #pragma unroll
                for (int q = 0; q < Kk; ++q)
                    if (q == minidx) { vals[q] = sc; adjs[q] = aj; cols[q] = c; }
                curmin = vals[0]; minidx = 0;
                #pragma unroll
                for (int q = 1; q < Kk; ++q)
                    if (vals[q] < curmin) { curmin = vals[q]; minidx = q; }
            }
        }

        // Sort kept columns ascending (torch nonzero row-major order).
        #pragma unroll
        for (int i = 0; i < Kk; ++i) { mCol[lane][i] = cols[i]; mAdj[lane][i] = adjs[i]; }
        for (int i = 1; i < Kk; ++i) {
            int ci = mCol[lane][i]; float ai = mAdj[lane][i];
            int j = i - 1;
            while (j >= 0 && mCol[lane][j] > ci) {
                mCol[lane][j + 1] = mCol[lane][j];
                mAdj[lane][j + 1] = mAdj[lane][j];
                --j;
            }
            mCol[lane][j + 1] = ci; mAdj[lane][j + 1] = ai;
        }

        // Emit [src | dst | HEW] as write-once nontemporal stores.
        #pragma unroll
        for (int k = 0; k < Kk; ++k) {
            int   c = mCol[lane][k];
            float w = mAdj[lane][k];
            #pragma unroll
            for (int b = 0; b < Bsz; ++b) {
                size_t o = (size_t)b * Nn * Kk + (size_t)row * Kk + k;
                __builtin_nontemporal_store((float)(row + b * Nn), &out[o]);
                __builtin_nontemporal_store((float)(c + b * Mm),   &out[BNK + o]);
                __builtin_nontemporal_store(w,                     &out[2 * BNK + o]);
            }
        }
    }
}

extern "C" void kernel_launch(void* const* d_in, const int* in_sizes, int n_in,
                              void* d_out, int out_size, void* d_ws, size_t ws_size,
                              hipStream_t stream) {
    // d_in[0] = x (unused beyond batch dim), d_in[1]=nodevec, d_in[2]=edgevec, d_in[3]=noise
    const float* nodevec = (const float*)d_in[1];
    const float* edgevec = (const float*)d_in[2];
    const float* noise   = (const float*)d_in[3];

    float* DE  = (float*)d_ws;           // 400,000 floats
    float* EE  = DE + Nn * Dd;           // 160,000 floats
    float* out = (float*)d_out;

    const int prepTotal = Nn * Dd;       // covers EE too (smaller)
    hgcn_prep<<<(prepTotal + 255) / 256, 256, 0, stream>>>(nodevec, edgevec, DE, EE);
    hgcn_topk<<<Nn / 16, 32 * NW, 0, stream>>>(DE, EE, noise, out);
}